// EdgeGAT_44538810859689
// MI455X (gfx1250) — compile-verified
//
#include <hip/hip_runtime.h>
#include <hip/hip_bf16.h>

#define DD 128
#define DE 64
#define LAMDA 0.5f
#define NEG_SLOPE 0.01f

typedef __attribute__((ext_vector_type(2))) float v2f;
typedef __attribute__((ext_vector_type(8))) float v8f;

// ---------------------------------------------------------------------------
// k0: zero h (N*D), init m_key to -inf key, denom to 0
// ---------------------------------------------------------------------------
__global__ void k_init(float* __restrict__ h, int* __restrict__ m_key,
                       float* __restrict__ denom, int n, int total) {
    int i = blockIdx.x * blockDim.x + threadIdx.x;
    if (i < total) h[i] = 0.0f;
    if (i < n) { m_key[i] = (int)0x80000000; denom[i] = 0.0f; }
}

// ---------------------------------------------------------------------------
// k1: fold weights.
//   u_s[k] = sum_j fc_w[j,k] * w_s[j],  u_d likewise (128 each)
//   v_e[k] = sum_j edge_fc_w[j,k] * w_e[j]  (64)
//   c_e    = edge_fc_b . w_e  (scalar)
// ---------------------------------------------------------------------------
__global__ void k_fold(const float* __restrict__ fc_w,
                       const float* __restrict__ edge_fc_w,
                       const float* __restrict__ edge_fc_b,
                       const float* __restrict__ attn_w,
                       float* __restrict__ u_s, float* __restrict__ u_d,
                       float* __restrict__ v_e, float* __restrict__ c_e) {
    int k = threadIdx.x;                       // 0..127, one block
    float ss = 0.0f, dd = 0.0f;
    for (int j = 0; j < DD; ++j) {
        float w = fc_w[j * DD + k];
        ss += w * attn_w[j];
        dd += w * attn_w[DD + j];
    }
    u_s[k] = ss; u_d[k] = dd;
    if (k < DE) {
        float ee = 0.0f;
        for (int j = 0; j < DE; ++j) ee += edge_fc_w[j * DE + k] * attn_w[2 * DD + j];
        v_e[k] = ee;
    }
    if (k == 0) {
        float cc = 0.0f;
        for (int j = 0; j < DE; ++j) cc += edge_fc_b[j] * attn_w[2 * DD + j];
        *c_e = cc;
    }
}

// ---------------------------------------------------------------------------
// k2: z = node_h @ fc_w.T via V_WMMA_F32_16X16X4_F32.
// Block = 256 threads = 8 waves. Block owns a 16-row tile; wave w owns
// columns [16w, 16w+16). fc_w (64 KB) staged in LDS once per block.
// A fragment (16x4 f32, 2 VGPRs): lane l -> M = l&15, K = 2*(l>>4) + {0,1}
// B fragment (4x16 f32, 2 VGPRs): lane l -> N = l&15, K = 2*(l>>4) + {0,1}
// D (16x16 f32, 8 VGPRs): vgpr r -> M = r + 8*(l>>4), N = l&15
// ---------------------------------------------------------------------------
__global__ __launch_bounds__(256) void k_node_gemm(
        const float* __restrict__ node_h, const float* __restrict__ fc_w,
        float* __restrict__ z, int n) {
    __shared__ float w_lds[DD * DD];           // 64 KB of 320 KB WGP LDS
    int t = threadIdx.x;
    int row0 = blockIdx.x * 16;
    if (row0 + 16 > n) return;                 // uniform per block
    for (int i = t; i < DD * DD; i += 256) w_lds[i] = fc_w[i];
    __syncthreads();

    int wave = t >> 5;                         // 0..7 -> column tile
    int lane = t & 31;
    int col0 = wave * 16;
    int mn   = lane & 15;                      // M for A, N for B/D
    int kh   = (lane >> 4) * 2;                // K sub-offset per lane half

    const float* arow = node_h + (size_t)(row0 + mn) * DD;
    const float* brow = w_lds + (col0 + mn) * DD;   // fc_w[col][k] (= B^T row)

    v8f acc = {0.f, 0.f, 0.f, 0.f, 0.f, 0.f, 0.f, 0.f};
    #pragma unroll 4
    for (int kk = 0; kk < DD; kk += 4) {
        int k = kk + kh;
        v2f a; a.x = arow[k]; a.y = arow[k + 1];
        v2f b; b.x = brow[k]; b.y = brow[k + 1];
        acc = __builtin_amdgcn_wmma_f32_16x16x4_f32(
            /*neg_a=*/false, a, /*neg_b=*/false, b,
            /*c_mod=*/(short)0, acc, /*reuse_a=*/false, /*reuse_b=*/false);
    }

    int mbase = (lane >> 4) * 8;
    float* zp = z + (size_t)row0 * DD + col0 + mn;
    #pragma unroll
    for (int r = 0; r < 8; ++r)
        zp[(size_t)(mbase + r) * DD] = acc[r];
}

// ---------------------------------------------------------------------------
// k3: s_node[i] = node_h[i] . u_s ; d_node[i] = node_h[i] . u_d
// One wave32 per node; lane handles k, k+32, k+64, k+96.
// ---------------------------------------------------------------------------
__global__ __launch_bounds__(256) void k_node_sd(
        const float* __restrict__ node_h, const float* __restrict__ u_s,
        const float* __restrict__ u_d, float* __restrict__ s_node,
        float* __restrict__ d_node, int n) {
    int node = blockIdx.x * 8 + (threadIdx.x >> 5);
    int lane = threadIdx.x & 31;
    if (node >= n) return;
    const float* row = node_h + (size_t)node * DD;
    float ss = 0.0f, dd = 0.0f;
    #pragma unroll
    for (int c = 0; c < 4; ++c) {
        int k = lane + 32 * c;
        float x = row[k];
        ss += x * u_s[k];
        dd += x * u_d[k];
    }
    #pragma unroll
    for (int off = 16; off; off >>= 1) {
        ss += __shfl_xor(ss, off, 32);
        dd += __shfl_xor(dd, off, 32);
    }
    if (lane == 0) { s_node[node] = ss; d_node[node] = dd; }
}

// ---------------------------------------------------------------------------
// k4: per-edge score e = lrelu(s[src] + d[dst] + edge_h . v_e + c_e),
// plus segment-max via monotone int-key atomicMax.
// One wave32 per edge (64-wide dot, 2 elems/lane).
// ---------------------------------------------------------------------------
__global__ __launch_bounds__(256) void k_edge_score(
        const float* __restrict__ edge_h, const int* __restrict__ src,
        const int* __restrict__ dst, const float* __restrict__ s_node,
        const float* __restrict__ d_node, const float* __restrict__ v_e,
        const float* __restrict__ c_e, float* __restrict__ e_lr,
        int* __restrict__ m_key, long long ecnt) {
    long long edge = (long long)blockIdx.x * 8 + (threadIdx.x >> 5);
    int lane = threadIdx.x & 31;
    if (edge >= ecnt) return;
    const float* row = edge_h + edge * (long long)DE;
    __builtin_prefetch(row + 8 * DE, 0, 0);    // stream next rows into cache
    float acc = row[lane] * v_e[lane] + row[lane + 32] * v_e[lane + 32];
    #pragma unroll
    for (int off = 16; off; off >>= 1) acc += __shfl_xor(acc, off, 32);
    if (lane == 0) {
        float ev = acc + *c_e + s_node[src[edge]] + d_node[dst[edge]];
        ev = ev > 0.0f ? ev : ev * NEG_SLOPE;
        e_lr[edge] = ev;
        int bits = __float_as_int(ev);
        int key = bits >= 0 ? bits : (bits ^ 0x7fffffff);  // monotone map
        atomicMax(&m_key[dst[edge]], key);
    }
}

// ---------------------------------------------------------------------------
// k5: ex = exp(e - m[dst]) ; denom[dst] += ex   (thread per edge)
// ---------------------------------------------------------------------------
__global__ void k_softmax_num(const float* __restrict__ e_lr,
                              const int* __restrict__ dst,
                              const int* __restrict__ m_key,
                              float* __restrict__ ex,
                              float* __restrict__ denom, long long ecnt) {
    long long i = (long long)blockIdx.x * blockDim.x + threadIdx.x;
    if (i >= ecnt) return;
    int d = dst[i];
    int key = m_key[d];
    int bits = key >= 0 ? key : (key ^ 0x7fffffff);
    float m = __int_as_float(bits);
    float v = __expf(e_lr[i] - m);
    ex[i] = v;
    atomicAdd(&denom[d], v);
}

// ---------------------------------------------------------------------------
// k6: h[dst] += (ex/denom[dst]) * z[src]   (wave per edge, 4 cols/lane)
// z (20 MB) lives in the 192 MB L2, so the gather is cheap.
// ---------------------------------------------------------------------------
__global__ __launch_bounds__(256) void k_scatter(
        const float* __restrict__ z, const int* __restrict__ src,
        const int* __restrict__ dst, const float* __restrict__ ex,
        const float* __restrict__ denom, float* __restrict__ h,
        long long ecnt) {
    long long edge = (long long)blockIdx.x * 8 + (threadIdx.x >> 5);
    int lane = threadIdx.x & 31;
    if (edge >= ecnt) return;
    int d = dst[edge];
    float alpha = ex[edge] / denom[d];
    const float* zr = z + (size_t)src[edge] * DD;
    float* hr = h + (size_t)d * DD;
    #pragma unroll
    for (int c = 0; c < 4; ++c) {
        int col = lane + 32 * c;
        atomicAdd(&hr[col], alpha * zr[col]);
    }
}

// ---------------------------------------------------------------------------
// k7: out = (1-lambda)*node_h + lambda*h
// ---------------------------------------------------------------------------
__global__ void k_final(const float* __restrict__ node_h,
                        const float* __restrict__ h,
                        float* __restrict__ out, int total) {
    int i = blockIdx.x * blockDim.x + threadIdx.x;
    if (i < total) out[i] = (1.0f - LAMDA) * node_h[i] + LAMDA * h[i];
}

// ---------------------------------------------------------------------------
extern "C" void kernel_launch(void* const* d_in, const int* in_sizes, int n_in,
                              void* d_out, int out_size, void* d_ws, size_t ws_size,
                              hipStream_t stream) {
    const float* node_h    = (const float*)d_in[0];
    const float* edge_h    = (const float*)d_in[1];
    const int*   src       = (const int*)  d_in[2];
    const int*   dst       = (const int*)  d_in[3];
    const float* fc_w      = (const float*)d_in[4];
    const float* edge_fc_w = (const float*)d_in[5];
    const float* edge_fc_b = (const float*)d_in[6];
    const float* attn_w    = (const float*)d_in[7];
    float* out = (float*)d_out;

    const int       n    = in_sizes[0] / DD;
    const long long ecnt = (long long)in_sizes[2];
    const int total = n * DD;

    // carve workspace (256B aligned)
    char* p = (char*)d_ws;
    auto carve = [&](size_t bytes) -> void* {
        void* r = (void*)p;
        p += (bytes + 255) & ~(size_t)255;
        return r;
    };
    float* z      = (float*)carve((size_t)total * sizeof(float));      // 20.5 MB
    float* h      = (float*)carve((size_t)total * sizeof(float));      // 20.5 MB
    float* s_node = (float*)carve((size_t)n * sizeof(float));
    float* d_node = (float*)carve((size_t)n * sizeof(float));
    float* denom  = (float*)carve((size_t)n * sizeof(float));
    int*   m_key  = (int*)  carve((size_t)n * sizeof(int));
    float* e_lr   = (float*)carve((size_t)ecnt * sizeof(float));       // 2.6 MB
    float* ex     = (float*)carve((size_t)ecnt * sizeof(float));       // 2.6 MB
    float* u_s    = (float*)carve(DD * sizeof(float));
    float* u_d    = (float*)carve(DD * sizeof(float));
    float* v_e    = (float*)carve(DE * sizeof(float));
    float* c_e    = (float*)carve(sizeof(float));

    const int B = 256;
    k_init<<<(total + B - 1) / B, B, 0, stream>>>(h, m_key, denom, n, total);
    k_fold<<<1, DD, 0, stream>>>(fc_w, edge_fc_w, edge_fc_b, attn_w, u_s, u_d, v_e, c_e);
    k_node_gemm<<<(n + 15) / 16, B, 0, stream>>>(node_h, fc_w, z, n);
    k_node_sd<<<(n + 7) / 8, B, 0, stream>>>(node_h, u_s, u_d, s_node, d_node, n);
    k_edge_score<<<(unsigned)((ecnt + 7) / 8), B, 0, stream>>>(
        edge_h, src, dst, s_node, d_node, v_e, c_e, e_lr, m_key, ecnt);
    k_softmax_num<<<(unsigned)((ecnt + B - 1) / B), B, 0, stream>>>(
        e_lr, dst, m_key, ex, denom, ecnt);
    k_scatter<<<(unsigned)((ecnt + 7) / 8), B, 0, stream>>>(
        z, src, dst, ex, denom, h, ecnt);
    k_final<<<(total + B - 1) / B, B, 0, stream>>>(node_h, h, out, total);
}